// Caps_Layer_12773232738744
// MI455X (gfx1250) — compile-verified
//
#include <hip/hip_runtime.h>
#include <hip/hip_bf16.h>
#include <math.h>

typedef float v2f __attribute__((ext_vector_type(2)));
typedef float v8f __attribute__((ext_vector_type(8)));

#define BATCH 64
#define NIN   1024
#define DIN   256
#define NC    32
#define DC    32
#define JT    64          // input-capsule columns per route workgroup
#define XS_STRIDE 260     // 256 + 4 floats pad: 16B aligned rows, bank-spread
#define LS_STRIDE 68      // 64 + 4
#define EPS   1e-7f

// ---------------------------------------------------------------- zero
__global__ __launch_bounds__(256) void zero_kernel(float* __restrict__ p, int n) {
    int i = blockIdx.x * 256 + threadIdx.x;
    if (i < n) p[i] = 0.0f;
}

// ---------------------------------------------------------------- colsum: S[b][d] = sum_j x[b][j][d]
__global__ __launch_bounds__(256) void colsum_kernel(const float* __restrict__ x,
                                                     float* __restrict__ S) {
    const int b = blockIdx.x;
    const int jc = blockIdx.y;           // 8 chunks of 128 rows
    const int d = threadIdx.x;           // 256 columns
    const float* xp = x + ((size_t)b * NIN + (size_t)jc * 128) * DIN + d;
    float s = 0.0f;
#pragma unroll 4
    for (int r = 0; r < 128; ++r) s += xp[(size_t)r * DIN];
    atomicAdd(&S[b * DIN + d], s);
}

// ---------------------------------------------------------------- outputs + squash + Z
// mode 0: Ybar[i][d] = S[b][d]/32 (round 0, uniform c) -> write Z
// mode 1: Ybar = Y[b]                                  -> write Z
// mode 2: Ybar = Y[b]                                  -> write final outputs to d_out
__global__ __launch_bounds__(256) void outz_kernel(const float* __restrict__ S,
                                                   const float* __restrict__ Y,
                                                   const float* __restrict__ W,
                                                   float* __restrict__ Z,
                                                   float* __restrict__ out,
                                                   int mode) {
    __shared__ float Yb[NC][XS_STRIDE];
    __shared__ float Os[NC][DC + 1];
    __shared__ float scale[NC];
    const int b = blockIdx.x;
    const int t = threadIdx.x;

    if (mode == 0) {
        Yb[0][t] = S[b * DIN + t] * (1.0f / 32.0f);   // t == d, blockDim==DIN
    } else {
        const float4* yg = reinterpret_cast<const float4*>(Y + (size_t)b * NC * DIN);
        for (int idx = t; idx < NC * (DIN / 4); idx += 256) {
            int r = idx >> 6, c4 = idx & 63;
            *reinterpret_cast<float4*>(&Yb[r][c4 * 4]) = yg[r * 64 + c4];
        }
    }
    __syncthreads();

    // o[i][k] = sum_d Ybar[i][d] * W[d][i*32+k]; 1024 (i,k) pairs, 4 per thread
    float o[4];
#pragma unroll
    for (int q = 0; q < 4; ++q) {
        const int p = t + 256 * q;
        const float* yrow = (mode == 0) ? Yb[0] : Yb[p >> 5];
        float s = 0.0f;
        for (int d = 0; d < DIN; ++d) s = fmaf(yrow[d], W[(size_t)d * 1024 + p], s);
        o[q] = s;
    }
#pragma unroll
    for (int q = 0; q < 4; ++q) { int p = t + 256 * q; Os[p >> 5][p & 31] = o[q]; }
    __syncthreads();

    if (t < NC) {   // squash scale per capsule: 1/sqrt(sum v^2 + eps)
        float s = 0.0f;
        for (int k = 0; k < DC; ++k) { float v = Os[t][k]; s = fmaf(v, v, s); }
        scale[t] = 1.0f / sqrtf(s + EPS);
    }
    __syncthreads();

    if (mode == 2) {
#pragma unroll
        for (int q = 0; q < 4; ++q) {
            int p = t + 256 * q; int i = p >> 5;
            out[(size_t)b * NC * DC + p] = Os[i][p & 31] * scale[i];
        }
        return;
    }

#pragma unroll
    for (int q = 0; q < 4; ++q) { int p = t + 256 * q; int i = p >> 5; Os[i][p & 31] *= scale[i]; }
    __syncthreads();

    // Z[i][d] = sum_k o_hat[i][k] * W[d][i*32+k]; thread t == d
    float* Zg = Z + (size_t)b * NC * DIN;
    for (int i = 0; i < NC; ++i) {
        const float* wp = W + (size_t)t * 1024 + i * 32;
        float s = 0.0f;
#pragma unroll
        for (int k = 0; k < DC; ++k) s = fmaf(Os[i][k], wp[k], s);
        Zg[i * DIN + t] = s;
    }
}

// ---------------------------------------------------------------- routing round (heavy, WMMA)
// Per (batch, j-chunk): L = Z @ X^T (M32,N64,K256); C = softmax_i(L);
// Y[b] += C @ X (M32,N256,K64) via global fp32 atomics.
__global__ __launch_bounds__(256) void route_kernel(const float* __restrict__ x,
                                                    const float* __restrict__ Z,
                                                    float* __restrict__ Y) {
    extern __shared__ float lds[];
    float* Xs = lds;                       // [JT][XS_STRIDE]
    float* Zs = Xs + JT * XS_STRIDE;       // [NC][XS_STRIDE]
    float* Ls = Zs + NC * XS_STRIDE;       // [NC][LS_STRIDE]

    const int b    = blockIdx.x;
    const int j0   = blockIdx.y * JT;
    const int t    = threadIdx.x;
    const int lane = t & 31;
    const int wave = t >> 5;               // 8 waves
    const int hi   = lane >> 4;            // K-half select per f32 WMMA layout
    const int l15  = lane & 15;

    // ---- stage X chunk [JT][256] and Z[b] [32][256] into padded LDS (coalesced b128)
    {
        const float4* xg = reinterpret_cast<const float4*>(x + ((size_t)b * NIN + j0) * DIN);
        for (int idx = t; idx < JT * (DIN / 4); idx += 256) {
            int r = idx >> 6, c4 = idx & 63;
            *reinterpret_cast<float4*>(&Xs[r * XS_STRIDE + c4 * 4]) = xg[r * 64 + c4];
        }
        const float4* zg = reinterpret_cast<const float4*>(Z + (size_t)b * NC * DIN);
        for (int idx = t; idx < NC * (DIN / 4); idx += 256) {
            int r = idx >> 6, c4 = idx & 63;
            *reinterpret_cast<float4*>(&Zs[r * XS_STRIDE + c4 * 4]) = zg[r * 64 + c4];
        }
    }
    __syncthreads();

    // ---- GEMM1: L[i][j] = sum_d Z[i][d] * X[j][d].  8 tiles of 16x16, one per wave.
    {
        const int mt = wave >> 2, nt = wave & 3;
        const int am = mt * 16 + l15;      // A row (capsule i)
        const int bn = nt * 16 + l15;      // B col (input capsule j), B[d][j] = X[j][d]
        v8f acc = {};
        for (int d0 = 0; d0 < DIN; d0 += 4) {
            const int kb = d0 + hi * 2;    // lanes 0-15: K=d0,d0+1; lanes 16-31: K=d0+2,d0+3
            v2f a  = *reinterpret_cast<const v2f*>(&Zs[am * XS_STRIDE + kb]);
            v2f bb = *reinterpret_cast<const v2f*>(&Xs[bn * XS_STRIDE + kb]);
            acc = __builtin_amdgcn_wmma_f32_16x16x4_f32(false, a, false, bb,
                                                        (short)0, acc, false, false);
        }
#pragma unroll
        for (int v = 0; v < 8; ++v)        // D: VGPR v -> M=v (lanes 0-15), M=v+8 (16-31)
            Ls[(mt * 16 + v + 8 * hi) * LS_STRIDE + nt * 16 + l15] = acc[v];
    }
    __syncthreads();

    // ---- softmax over the 32 capsules per column j (matches jax softmax axis=1)
    if (t < JT) {
        float m = -3.4e38f;
        for (int i = 0; i < NC; ++i) m = fmaxf(m, Ls[i * LS_STRIDE + t]);
        float s = 0.0f;
        for (int i = 0; i < NC; ++i) {
            float e = __expf(Ls[i * LS_STRIDE + t] - m);
            Ls[i * LS_STRIDE + t] = e; s += e;
        }
        float r = 1.0f / s;
        for (int i = 0; i < NC; ++i) Ls[i * LS_STRIDE + t] *= r;
    }
    __syncthreads();

    // ---- GEMM2: Yp[i][d] = sum_j C[i][j] * X[j][d].  32 tiles, 4 per wave (shared A frag).
    {
        const int mt = wave >> 2;          // waves 0-3 -> M rows 0-15, waves 4-7 -> 16-31
        const int am = mt * 16 + l15;
        v8f acc[4] = {{}, {}, {}, {}};
        for (int jj = 0; jj < JT; jj += 4) {
            const int kb = jj + hi * 2;
            v2f a = *reinterpret_cast<const v2f*>(&Ls[am * LS_STRIDE + kb]);
#pragma unroll
            for (int q = 0; q < 4; ++q) {
                const int nt = (wave & 3) * 4 + q;
                const int bn = nt * 16 + l15;   // B[j][d] = X[j][d], row-major: two b32 loads
                v2f bb;
                bb.x = Xs[kb * XS_STRIDE + bn];
                bb.y = Xs[(kb + 1) * XS_STRIDE + bn];
                acc[q] = __builtin_amdgcn_wmma_f32_16x16x4_f32(false, a, false, bb,
                                                               (short)0, acc[q], false, false);
            }
        }
        float* Yg = Y + (size_t)b * NC * DIN;
#pragma unroll
        for (int q = 0; q < 4; ++q) {
            const int n = ((wave & 3) * 4 + q) * 16 + l15;
#pragma unroll
            for (int v = 0; v < 8; ++v) {
                const int m = mt * 16 + v + 8 * hi;
                atomicAdd(&Yg[m * DIN + n], acc[q][v]);   // global_atomic_add_f32
            }
        }
    }
}

// ---------------------------------------------------------------- launch
extern "C" void kernel_launch(void* const* d_in, const int* in_sizes, int n_in,
                              void* d_out, int out_size, void* d_ws, size_t ws_size,
                              hipStream_t stream) {
    const float* x = (const float*)d_in[0];      // [64,1024,256]
    const float* W = (const float*)d_in[1];      // [256,1024]
    float* out = (float*)d_out;                  // [64,32,32]
    float* ws  = (float*)d_ws;

    float* S = ws;                               // [64,256]
    float* Y = ws + BATCH * DIN;                 // [64,32,256]
    float* Z = Y + (size_t)BATCH * NC * DIN;     // [64,32,256]
    const int nSY = BATCH * DIN + BATCH * NC * DIN;
    const int nY  = BATCH * NC * DIN;

    const size_t routeShm = (size_t)(JT * XS_STRIDE + NC * XS_STRIDE + NC * LS_STRIDE)
                            * sizeof(float);     // ~106 KB (LDS is 320 KB/WGP on CDNA5)

    zero_kernel<<<(nSY + 255) / 256, 256, 0, stream>>>(S, nSY);
    colsum_kernel<<<dim3(BATCH, 8), 256, 0, stream>>>(x, S);
    outz_kernel<<<BATCH, 256, 0, stream>>>(S, Y, W, Z, out, 0);

    for (int r = 1; r <= 2; ++r) {
        zero_kernel<<<(nY + 255) / 256, 256, 0, stream>>>(Y, nY);
        route_kernel<<<dim3(BATCH, NIN / JT), 256, routeShm, stream>>>(x, Z, Y);
        outz_kernel<<<BATCH, 256, 0, stream>>>(S, Y, W, Z, out, r);
    }
}